// TextProposal_53953379172896
// MI455X (gfx1250) — compile-verified
//
#include <hip/hip_runtime.h>
#include <math.h>

#define BB   2
#define NN   100000
#define MM   60000
#define CC   2
#define OUTN 300
#define IOU_THR   0.3f
#define SCORE_THR 0.7f
#define NEG_INF (-__builtin_inff())

#define NMS_THREADS 1024
#define NWAVES (NMS_THREADS / 32)

typedef int v4i_ __attribute__((vector_size(16)));
typedef __attribute__((address_space(1))) v4i_* gv4p;
typedef __attribute__((address_space(3))) v4i_* lv4p;

// ---- gfx1250 async global->LDS path (probe-verified in rounds 1-2)
#if defined(__HIP_DEVICE_COMPILE__) && defined(__has_builtin)
#if __has_builtin(__builtin_amdgcn_global_load_async_to_lds_b128)
#define USE_ASYNC_LDS 1
#endif
#endif

// -----------------------------------------------------------------------------
// Phase 1: gather + softmax + box regression. One thread per (b, m).
// ws layout (floats):
//   boxes  [B*M*4]  (float4 per proposal: y1,x1,y2,x2)
//   scores [B*M]
//   work   [B*M]    (score if > 0.7 else -inf)
//   logits [B*M*2]
// -----------------------------------------------------------------------------
__global__ __launch_bounds__(256) void ctpn_prep_kernel(
    const float* __restrict__ deltas,        // [B,N,2]
    const float* __restrict__ class_logits,  // [B,N,2]
    const float* __restrict__ anchors,       // [B,M,4]
    const int*   __restrict__ vai,           // [B,M]
    float* __restrict__ boxes,
    float* __restrict__ scores,
    float* __restrict__ work,
    float* __restrict__ logits) {
  int g = blockIdx.x * blockDim.x + threadIdx.x;
  if (g >= BB * MM) return;
  int b = g / MM;

  float4 a = ((const float4*)anchors)[g];        // y1,x1,y2,x2
  int idx = vai[g];

  float2 d = ((const float2*)deltas)[(size_t)b * NN + idx];
  float dy = d.x * 0.1f;
  float dh = d.y * 0.2f;
  float h  = a.z - a.x;
  float cy = (a.x + a.z) * 0.5f;
  cy += dy * h;
  h  *= expf(dh);
  float4 bx = make_float4(cy - h * 0.5f, a.y, cy + h * 0.5f, a.w);
  ((float4*)boxes)[g] = bx;

  float2 l = ((const float2*)class_logits)[(size_t)b * NN + idx];
  float s = 1.0f / (1.0f + expf(l.x - l.y));    // softmax fg prob, C==2
  scores[g] = s;
  work[g]   = (s > SCORE_THR) ? s : NEG_INF;
  ((float2*)logits)[g] = l;
}

// -----------------------------------------------------------------------------
// Phase 2: greedy NMS. One workgroup per batch; the 60000-entry work array
// lives entirely in LDS (240 KB -- CDNA5's 320KB WGP LDS makes this possible).
// All LDS traffic on the 300-deep serial critical path uses ds_load_b128
// (float4) to minimize DS issue count per iteration.
// -----------------------------------------------------------------------------
__global__ __launch_bounds__(NMS_THREADS) void ctpn_nms_kernel(
    const float* __restrict__ boxes,    // [B*M*4]
    const float* __restrict__ scores,   // [B*M]
    const float* __restrict__ work_g,   // [B*M]
    const float* __restrict__ logits,   // [B*M*2]
    float* __restrict__ out_boxes,      // [B*OUT*5]
    float* __restrict__ out_scores,     // [B*OUT*2]
    float* __restrict__ out_logits) {   // [B*OUT*3]
  extern __shared__ float work[];       // MM floats, 16B aligned
  __shared__ float  rv[NWAVES];
  __shared__ int    ri[NWAVES];
  __shared__ int    sBest;
  __shared__ float4 sBox;

  const int b   = blockIdx.x;
  const int tid = threadIdx.x;
  const float*  wg = work_g + (size_t)b * MM;
  const float4* bx = (const float4*)boxes + (size_t)b * MM;
  float4* w4 = (float4*)work;
  const int M4 = MM / 4;                // 15000 float4 slots

  // ---- fill LDS work array via async DMA (ASYNCcnt-tracked, b128 granularity)
#ifdef USE_ASYNC_LDS
  {
    float* wgm = const_cast<float*>(wg);
    for (int j = tid * 4; j < MM; j += NMS_THREADS * 4) {
      __builtin_amdgcn_global_load_async_to_lds_b128(
          (gv4p)(v4i_*)(wgm + j),
          (lv4p)(v4i_*)(work + j),
          0, 0);
    }
  }
#if __has_builtin(__builtin_amdgcn_s_wait_asynccnt)
  __builtin_amdgcn_s_wait_asynccnt(0);
#else
  asm volatile("s_wait_asynccnt 0" ::: "memory");
#endif
#else
  for (int j = tid; j < MM; j += NMS_THREADS) work[j] = wg[j];
#endif
  __syncthreads();

  for (int it = 0; it < OUTN; ++it) {
    // ---- parallel argmax over LDS, b128 reads.
    // Ascending index order + strict '>' keeps first-occurrence semantics
    // (matches jnp.argmax); cross-thread ties resolve to lowest index below.
    float bv = NEG_INF;
    int   bi = 0x7FFFFFFF;
    for (int j4 = tid; j4 < M4; j4 += NMS_THREADS) {
      float4 v = w4[j4];
      int base = j4 * 4;
      if (v.x > bv) { bv = v.x; bi = base;     }
      if (v.y > bv) { bv = v.y; bi = base + 1; }
      if (v.z > bv) { bv = v.z; bi = base + 2; }
      if (v.w > bv) { bv = v.w; bi = base + 3; }
    }
#pragma unroll
    for (int off = 16; off > 0; off >>= 1) {
      float ov = __shfl_down(bv, off, 32);
      int   oi = __shfl_down(bi, off, 32);
      if (ov > bv || (ov == bv && oi < bi)) { bv = ov; bi = oi; }
    }
    if ((tid & 31) == 0) { rv[tid >> 5] = bv; ri[tid >> 5] = bi; }
    __syncthreads();
    if (tid < 32) {
      bv = rv[tid]; bi = ri[tid];
#pragma unroll
      for (int off = 16; off > 0; off >>= 1) {
        float ov = __shfl_down(bv, off, 32);
        int   oi = __shfl_down(bi, off, 32);
        if (ov > bv || (ov == bv && oi < bi)) { bv = ov; bi = oi; }
      }
      if (tid == 0) {
        if (bv > NEG_INF) {        // valid selection
          float4 bbb = bx[bi];
          sBox  = bbb;
          sBest = bi;
          int row = b * OUTN + it;
          float* ob = out_boxes + (size_t)row * 5;
          ob[0] = bbb.x; ob[1] = bbb.y; ob[2] = bbb.z; ob[3] = bbb.w; ob[4] = 1.0f;
          float* os = out_scores + (size_t)row * 2;
          os[0] = scores[(size_t)b * MM + bi]; os[1] = 1.0f;
          const float* lp = logits + ((size_t)b * MM + bi) * 2;
          float* ol = out_logits + (size_t)row * 3;
          ol[0] = lp[0]; ol[1] = lp[1]; ol[2] = 1.0f;
        } else {
          sBest = -1;
        }
      }
    }
    __syncthreads();

    int best = sBest;
    if (best < 0) {
      // all remaining rows are exact zeros (tag==0 in the reference)
      for (int r = it + tid; r < OUTN; r += NMS_THREADS) {
        int row = b * OUTN + r;
        float* ob = out_boxes  + (size_t)row * 5;
        float* os = out_scores + (size_t)row * 2;
        float* ol = out_logits + (size_t)row * 3;
        ob[0] = 0.f; ob[1] = 0.f; ob[2] = 0.f; ob[3] = 0.f; ob[4] = 0.f;
        os[0] = 0.f; os[1] = 0.f;
        ol[0] = 0.f; ol[1] = 0.f; ol[2] = 0.f;
      }
      return;
    }

    // ---- IoU suppression vs best box (self-IoU==1 removes 'best' too).
    // One b128 LDS read covers 4 candidates; dead quads skip all global traffic;
    // single b128 write-back only when something changed. Each float4 slot is
    // owned by exactly one thread within this barrier-delimited pass.
    float4 bbb = sBox;
    float barea = (bbb.z - bbb.x) * (bbb.w - bbb.y);
    for (int j4 = tid; j4 < M4; j4 += NMS_THREADS) {
      __builtin_prefetch(bx + (j4 + NMS_THREADS) * 4, 0, 1);  // global_prefetch_b8
      float4 v = w4[j4];
      if (!(v.x > NEG_INF) && !(v.y > NEG_INF) &&
          !(v.z > NEG_INF) && !(v.w > NEG_INF)) continue;     // whole quad dead
      int base = j4 * 4;
      bool changed = false;
      float* vp = (float*)&v;
#pragma unroll
      for (int k = 0; k < 4; ++k) {
        if (!(vp[k] > NEG_INF)) continue;                     // -inf -> -inf is identity
        float4 cb = bx[base + k];
        float iy1 = fmaxf(bbb.x, cb.x);
        float ix1 = fmaxf(bbb.y, cb.y);
        float iy2 = fminf(bbb.z, cb.z);
        float ix2 = fminf(bbb.w, cb.w);
        float inter = fmaxf(iy2 - iy1, 0.0f) * fmaxf(ix2 - ix1, 0.0f);
        float carea = (cb.z - cb.x) * (cb.w - cb.y);
        float iou = inter / (barea + carea - inter);
        if (iou > IOU_THR) { vp[k] = NEG_INF; changed = true; }
      }
      if (changed) w4[j4] = v;
    }
    __syncthreads();
  }
}

// -----------------------------------------------------------------------------
extern "C" void kernel_launch(void* const* d_in, const int* in_sizes, int n_in,
                              void* d_out, int out_size, void* d_ws, size_t ws_size,
                              hipStream_t stream) {
  const float* deltas       = (const float*)d_in[0];
  // d_in[1] = side_deltas: unused (USE_SIDE_REFINE == False -> x from anchors)
  const float* class_logits = (const float*)d_in[2];
  const float* anchors      = (const float*)d_in[3];
  const int*   vai          = (const int*)d_in[4];

  float* ws      = (float*)d_ws;
  float* boxes   = ws;                              // B*M*4
  float* scoresW = boxes   + (size_t)BB * MM * 4;   // B*M
  float* workW   = scoresW + (size_t)BB * MM;       // B*M
  float* logitsW = workW   + (size_t)BB * MM;       // B*M*2  (total ~3.84 MB)

  int total = BB * MM;
  ctpn_prep_kernel<<<(total + 255) / 256, 256, 0, stream>>>(
      deltas, class_logits, anchors, vai, boxes, scoresW, workW, logitsW);

  float* out_boxes  = (float*)d_out;                    // [B,OUT,5]
  float* out_scores = out_boxes  + (size_t)BB * OUTN * 5;  // [B,OUT,2]
  float* out_logits = out_scores + (size_t)BB * OUTN * 2;  // [B,OUT,3]

  size_t lds_bytes = (size_t)MM * sizeof(float);    // 240 KB, fits CDNA5 WGP LDS
  ctpn_nms_kernel<<<BB, NMS_THREADS, lds_bytes, stream>>>(
      boxes, scoresW, workW, logitsW, out_boxes, out_scores, out_logits);
}